// EncoderS5Network_70342974374462
// MI455X (gfx1250) — compile-verified
//
#include <hip/hip_runtime.h>
#include <hip/hip_bf16.h>
#include <math.h>

// Problem sizes (fixed by the reference)
#define Tt   1024
#define Bb   64
#define Hh   256
#define Pp   128
#define HID  1024
#define OUTD 512
#define TB   (Tt * Bb)           // 65536 rows

typedef float v2f __attribute__((ext_vector_type(2)));
typedef float v8f __attribute__((ext_vector_type(8)));

__device__ __forceinline__ float gelu_tanh(float v) {
    // jax.nn.gelu default (approximate=True)
    float c = 0.7978845608028654f; // sqrt(2/pi)
    float u = c * (v + 0.044715f * v * v * v);
    return 0.5f * v * (1.0f + tanhf(u));
}

// ---------------------------------------------------------------------------
// Prep: Lambda_bar, packed B_bar (2P x H, rows [re;im]) and packed C (H x 2P,
// cols [2*C_re, -2*C_im]) so both S5 GEMMs become single real GEMMs with K=256.
// ---------------------------------------------------------------------------
__global__ __launch_bounds__(256) void prep_kernel(
    const float* __restrict__ Lre, const float* __restrict__ Lim,
    const float* __restrict__ logstep,
    const float* __restrict__ Bre, const float* __restrict__ Bim,
    const float* __restrict__ Cre, const float* __restrict__ Cim,
    float* __restrict__ Bp,   // (2P, H) row-major
    float* __restrict__ Cc,   // (H, 2P) row-major
    float* __restrict__ lamre, float* __restrict__ lamim)
{
    int idx = blockIdx.x * 256 + threadIdx.x;
    if (idx >= Pp * Hh) return;
    int p = idx / Hh, h = idx % Hh;

    float lr = Lre[p], li = Lim[p];
    float st = __expf(logstep[p]);
    float er = __expf(lr * st);
    float lbr = er * __cosf(li * st);
    float lbi = er * __sinf(li * st);
    // coef = (Lambda_bar - 1) / Lambda
    float den = lr * lr + li * li;
    float cr = ((lbr - 1.0f) * lr + lbi * li) / den;
    float ci = (lbi * lr - (lbr - 1.0f) * li) / den;

    float br = Bre[idx], bi = Bim[idx];
    Bp[(size_t)p * Hh + h]        = cr * br - ci * bi;   // Re(B_bar)
    Bp[(size_t)(Pp + p) * Hh + h] = cr * bi + ci * br;   // Im(B_bar)

    // reinterpret idx as (h2, p2) for C packing
    int h2 = idx / Pp, p2 = idx % Pp;
    Cc[(size_t)h2 * (2 * Pp) + p2]      =  2.0f * Cre[idx];
    Cc[(size_t)h2 * (2 * Pp) + Pp + p2] = -2.0f * Cim[idx];

    if (h == 0) { lamre[p] = lbr; lamim[p] = lbi; }
}

// ---------------------------------------------------------------------------
// LayerNorm over H=256, one row per 256-thread block (8 wave32s)
// ---------------------------------------------------------------------------
__global__ __launch_bounds__(256) void ln_kernel(
    const float* __restrict__ obs, const float* __restrict__ gamma,
    const float* __restrict__ beta, float* __restrict__ xbuf)
{
    int row = blockIdx.x;
    int tid = threadIdx.x;
    float v = obs[(size_t)row * Hh + tid];
    float s = v, q = v * v;
    #pragma unroll
    for (int o = 16; o > 0; o >>= 1) {
        s += __shfl_xor(s, o, 32);
        q += __shfl_xor(q, o, 32);
    }
    __shared__ float ssum[8], ssq[8];
    if ((tid & 31) == 0) { ssum[tid >> 5] = s; ssq[tid >> 5] = q; }
    __syncthreads();
    float ts = 0.f, tq = 0.f;
    #pragma unroll
    for (int i = 0; i < 8; i++) { ts += ssum[i]; tq += ssq[i]; }
    float mu = ts * (1.0f / Hh);
    float var = tq * (1.0f / Hh) - mu * mu;
    float r = rsqrtf(var + 1e-6f);
    xbuf[(size_t)row * Hh + tid] = (v - mu) * r * gamma[tid] + beta[tid];
}

// ---------------------------------------------------------------------------
// Sequential S5 scan with resets. One thread per (b,p); xs overwrites Bu
// in-place. Writes final hidden state [re(B,P); im(B,P)] to hid_out.
// ---------------------------------------------------------------------------
__global__ __launch_bounds__(256) void scan_kernel(
    const float* __restrict__ hre0, const float* __restrict__ him0,
    const unsigned char* __restrict__ dones,
    const float* __restrict__ lamre, const float* __restrict__ lamim,
    float* __restrict__ xs, float* __restrict__ hid_out)
{
    int g = blockIdx.x * 256 + threadIdx.x;
    if (g >= Bb * Pp) return;
    int b = g / Pp, p = g % Pp;
    float lr = lamre[p], li = lamim[p];
    float hr = hre0[g], hi = him0[g];
    for (int t = 0; t < Tt; t++) {
        size_t base = ((size_t)t * Bb + b) * (2 * Pp) + p;
        float br = xs[base], bi = xs[base + Pp];
        if (dones[t * Bb + b]) {
            hr = br; hi = bi;
        } else {
            float nr = lr * hr - li * hi + br;
            float ni = lr * hi + li * hr + bi;
            hr = nr; hi = ni;
        }
        xs[base] = hr; xs[base + Pp] = hi;
    }
    hid_out[g] = hr;
    hid_out[Bb * Pp + g] = hi;
}

// ---------------------------------------------------------------------------
// WMMA fp32 GEMM.  C[M,N] = A[M,K] * B (+ epilogue).
//   BT = 0 : B stored K x N (ldb = N row stride)
//   BT = 1 : B stored N x K (ldb = K row stride)  -> C = A * B^T
// Block: 256 threads = 8 wave32, tile BM=128 x BN=128, BK=32.
// Wave grid 4(M) x 2(N); each wave: 32x64 via 2x4 V_WMMA_F32_16X16X4_F32.
// B tile stored k-pair interleaved (Bs[k/2][n][2]) so each B fragment is one
// aligned 8-byte LDS load. Register-pipelined: next K-tile's global loads are
// issued right after the barrier so they are in flight during the WMMA phase.
// EPI: 0 none | 1 relu(acc+bias) | 2 skip+gelu(acc + x*D) | 3 acc+bias | 4 C+=acc
// ---------------------------------------------------------------------------
template <int BT, int EPI>
__global__ __launch_bounds__(256) void gemm_wmma(
    const float* __restrict__ A, const float* __restrict__ Bm,
    float* __restrict__ C, int M, int N, int K,
    int lda, int ldb, int ldc,
    const float* __restrict__ bias, const float* __restrict__ skip,
    const float* __restrict__ xrow, const float* __restrict__ Dvec)
{
    constexpr int BM = 128, BN = 128, BK = 32;
    constexpr int BNP = BN + 2;            // pad rows of the pair-layout
    __shared__ float As[BM][BK];           // 16 KB, rows 16B-aligned
    __shared__ float Bs[BK / 2][BNP][2];   // ~16.6 KB, (k-pair, n, k&1)

    const int tid = threadIdx.x;
    const int m0 = blockIdx.y * BM;
    const int n0 = blockIdx.x * BN;
    const int wid = tid >> 5;
    const int lane = tid & 31;
    const int wm = (wid >> 1) * 32;        // 0 / 32 / 64 / 96
    const int wn = (wid & 1) * 64;         // 0 / 64
    const int l15 = lane & 15;
    const int lh = lane >> 4;

    // fixed per-thread staging coordinates
    const int ar_[1] = {0};  (void)ar_;
    const int aR  = tid >> 3;              // A row for idx4 = tid (+ i*256 adds 32 rows)
    const int aC4 = (tid & 7) * 4;         // A col
    const int bR  = tid >> 5;              // B row  (BT=0; + i*256 adds 8 rows)
    const int bC4 = (tid & 31) * 4;        // B col  (BT=0)
    const int bNr = tid >> 3;              // B n-row (BT=1; + i*256 adds 32 rows)
    const int bK4 = (tid & 7) * 4;         // B k-col (BT=1)

    v8f acc[2][4] = {};
    float4 ra[4], rb[4];

    // prefetch tile 0 (batched: all 8 b128 loads in flight together)
    #pragma unroll
    for (int i = 0; i < 4; i++)
        ra[i] = *reinterpret_cast<const float4*>(
            &A[(size_t)(m0 + aR + i * 32) * lda + aC4]);
    #pragma unroll
    for (int i = 0; i < 4; i++) {
        if (BT == 0)
            rb[i] = *reinterpret_cast<const float4*>(
                &Bm[(size_t)(bR + i * 8) * ldb + n0 + bC4]);
        else
            rb[i] = *reinterpret_cast<const float4*>(
                &Bm[(size_t)(n0 + bNr + i * 32) * ldb + bK4]);
    }

    for (int k0 = 0; k0 < K; k0 += BK) {
        // --- drain prefetch registers into LDS ---
        #pragma unroll
        for (int i = 0; i < 4; i++)
            *reinterpret_cast<float4*>(&As[aR + i * 32][aC4]) = ra[i];
        if (BT == 0) {
            #pragma unroll
            for (int i = 0; i < 4; i++) {
                int r = bR + i * 8;
                int pr = r >> 1, sub = r & 1;
                Bs[pr][bC4 + 0][sub] = rb[i].x;
                Bs[pr][bC4 + 1][sub] = rb[i].y;
                Bs[pr][bC4 + 2][sub] = rb[i].z;
                Bs[pr][bC4 + 3][sub] = rb[i].w;
            }
        } else {
            #pragma unroll
            for (int i = 0; i < 4; i++) {
                int nr = bNr + i * 32;
                float2 lo = make_float2(rb[i].x, rb[i].y);
                float2 hi = make_float2(rb[i].z, rb[i].w);
                *reinterpret_cast<float2*>(&Bs[(bK4 >> 1) + 0][nr][0]) = lo;
                *reinterpret_cast<float2*>(&Bs[(bK4 >> 1) + 1][nr][0]) = hi;
            }
        }
        __syncthreads();

        // --- issue next tile's global loads; they overlap the WMMA phase ---
        const int kn = k0 + BK;
        if (kn < K) {
            #pragma unroll
            for (int i = 0; i < 4; i++)
                ra[i] = *reinterpret_cast<const float4*>(
                    &A[(size_t)(m0 + aR + i * 32) * lda + kn + aC4]);
            #pragma unroll
            for (int i = 0; i < 4; i++) {
                if (BT == 0)
                    rb[i] = *reinterpret_cast<const float4*>(
                        &Bm[(size_t)(kn + bR + i * 8) * ldb + n0 + bC4]);
                else
                    rb[i] = *reinterpret_cast<const float4*>(
                        &Bm[(size_t)(n0 + bNr + i * 32) * ldb + kn + bK4]);
            }
        }

        // --- compute: 8 k-steps; per step 8 WMMAs from 6 b64 LDS loads ---
        #pragma unroll
        for (int kk = 0; kk < BK; kk += 4) {
            const int kb = kk + 2 * lh;        // k, k+1 for this lane half
            const int kp = (kk >> 1) + lh;     // pair row in Bs
            v2f a[2], b[4];
            #pragma unroll
            for (int i = 0; i < 2; i++)        // A frag: lane = m + 16*(k>>1)
                a[i] = *reinterpret_cast<const v2f*>(&As[wm + i * 16 + l15][kb]);
            #pragma unroll
            for (int j = 0; j < 4; j++)        // B frag: lane = n + 16*(k>>1)
                b[j] = *reinterpret_cast<const v2f*>(&Bs[kp][wn + j * 16 + l15][0]);
            #pragma unroll
            for (int i = 0; i < 2; i++)
                #pragma unroll
                for (int j = 0; j < 4; j++)
                    acc[i][j] = __builtin_amdgcn_wmma_f32_16x16x4_f32(
                        false, a[i], false, b[j], (short)0, acc[i][j],
                        false, false);
        }
        __syncthreads();
    }

    // --- epilogue: vgpr r -> row (r + 8*lh), col l15 within each 16x16 tile ---
    #pragma unroll
    for (int i = 0; i < 2; i++) {
        #pragma unroll
        for (int j = 0; j < 4; j++) {
            #pragma unroll
            for (int r = 0; r < 8; r++) {
                int row = m0 + wm + i * 16 + lh * 8 + r;
                int col = n0 + wn + j * 16 + l15;
                size_t cidx = (size_t)row * ldc + col;
                float v = acc[i][j][r];
                if (EPI == 0) {
                    C[cidx] = v;
                } else if (EPI == 1) {
                    v += bias[col];
                    C[cidx] = fmaxf(v, 0.0f);
                } else if (EPI == 2) {
                    float v2 = v + xrow[cidx] * Dvec[col];
                    C[cidx] = skip[cidx] + gelu_tanh(v2);
                } else if (EPI == 3) {
                    C[cidx] = v + bias[col];
                } else { // EPI == 4
                    C[cidx] += v;
                }
            }
        }
    }
}

// ---------------------------------------------------------------------------
extern "C" void kernel_launch(void* const* d_in, const int* in_sizes, int n_in,
                              void* d_out, int out_size, void* d_ws, size_t ws_size,
                              hipStream_t stream) {
    const float* hidden_re = (const float*)d_in[0];
    const float* hidden_im = (const float*)d_in[1];
    const float* obs       = (const float*)d_in[2];
    const unsigned char* dones = (const unsigned char*)d_in[3]; // jnp bool_
    const float* Lambda_re = (const float*)d_in[4];
    const float* Lambda_im = (const float*)d_in[5];
    const float* B_re      = (const float*)d_in[6];
    const float* B_im      = (const float*)d_in[7];
    const float* C_re      = (const float*)d_in[8];
    const float* C_im      = (const float*)d_in[9];
    const float* Dv        = (const float*)d_in[10];
    const float* log_step  = (const float*)d_in[11];
    const float* ln_gamma  = (const float*)d_in[12];
    const float* ln_beta   = (const float*)d_in[13];
    const float* W1        = (const float*)d_in[14];
    const float* b1        = (const float*)d_in[15];
    const float* W2        = (const float*)d_in[16];
    const float* b2        = (const float*)d_in[17];

    float* hid_out = (float*)d_out;                 // (2,B,P)
    float* e_out   = (float*)d_out + 2 * Bb * Pp;   // (T,B,OUT)

    // workspace layout
    float* wsf   = (float*)d_ws;
    size_t off   = 0;
    float* Bp    = wsf + off; off += (size_t)2 * Pp * Hh;   // 65536
    float* Cc    = wsf + off; off += (size_t)Hh * 2 * Pp;   // 65536
    float* lamre = wsf + off; off += Pp;
    float* lamim = wsf + off; off += Pp;
    float* xbuf  = wsf + off; off += (size_t)TB * Hh;       // 64 MB
    float* bu    = wsf + off; off += (size_t)TB * 2 * Pp;   // 64 MB (Bu -> xs -> mlp tmp)
    float* ybuf  = wsf + off; off += (size_t)TB * Hh;       // 64 MB
    float* tbuf  = bu;  // reused after the C-projection GEMM consumed xs

    // 1) discretization + weight packing
    prep_kernel<<<(Pp * Hh + 255) / 256, 256, 0, stream>>>(
        Lambda_re, Lambda_im, log_step, B_re, B_im, C_re, C_im,
        Bp, Cc, lamre, lamim);

    // 2) layernorm
    ln_kernel<<<TB, 256, 0, stream>>>(obs, ln_gamma, ln_beta, xbuf);

    // 3) Bu = x @ B_bar^T  (packed re/im, N = 2P = 256, K = H = 256)
    gemm_wmma<1, 0><<<dim3(2, TB / 128), 256, 0, stream>>>(
        xbuf, Bp, bu, TB, 2 * Pp, Hh, Hh, Hh, 2 * Pp,
        nullptr, nullptr, nullptr, nullptr);

    // 4) sequential scan with resets (in-place on bu)
    scan_kernel<<<(Bb * Pp + 255) / 256, 256, 0, stream>>>(
        hidden_re, hidden_im, dones, lamre, lamim, bu, hid_out);

    // 5) y = obs + gelu(2*Re(xs @ C^T) + x*D)   (N = H = 256, K = 2P = 256)
    gemm_wmma<1, 2><<<dim3(2, TB / 128), 256, 0, stream>>>(
        bu, Cc, ybuf, TB, Hh, 2 * Pp, 2 * Pp, 2 * Pp, Hh,
        nullptr, obs, xbuf, Dv);

    // 6) MLP, chunked over HID in 4 slices of 256 to bound workspace:
    //    t = relu(y @ W1[:,c] + b1[c]);  e (+)= t @ W2[c,:]  (+ b2 on c==0)
    for (int c = 0; c < 4; c++) {
        gemm_wmma<0, 1><<<dim3(2, TB / 128), 256, 0, stream>>>(
            ybuf, W1 + c * 256, tbuf, TB, 256, Hh, Hh, HID, 256,
            b1 + c * 256, nullptr, nullptr, nullptr);
        if (c == 0) {
            gemm_wmma<0, 3><<<dim3(4, TB / 128), 256, 0, stream>>>(
                tbuf, W2 + (size_t)c * 256 * OUTD, e_out, TB, OUTD, 256,
                256, OUTD, OUTD, b2, nullptr, nullptr, nullptr);
        } else {
            gemm_wmma<0, 4><<<dim3(4, TB / 128), 256, 0, stream>>>(
                tbuf, W2 + (size_t)c * 256 * OUTD, e_out, TB, OUTD, 256,
                256, OUTD, OUTD, nullptr, nullptr, nullptr, nullptr);
        }
    }
}